// Pillars_to_PseudoIMG_22342419873901
// MI455X (gfx1250) — compile-verified
//
#include <hip/hip_runtime.h>
#include <hip/hip_bf16.h>

#define XSIZE 432
#define YSIZE 432
#define FDIM  64

// ---------------------------------------------------------------------------
// Kernel 1: zero the [B, F, X, Y] output with wide b128 stores.
// 191 MB of pure streaming writes -> runs at HBM rate (~8.2 us @ 23.3 TB/s).
// ---------------------------------------------------------------------------
__global__ void __launch_bounds__(256)
pseudoimg_zero_kernel(float4* __restrict__ out, long n4) {
    long i = (long)blockIdx.x * blockDim.x + threadIdx.x;
    if (i < n4) {
        out[i] = make_float4(0.f, 0.f, 0.f, 0.f);   // global_store_b128
    }
}

// ---------------------------------------------------------------------------
// Kernel 2: scatter-add pillar features into the pseudo-image.
// Grid = (P*64/256, B): batch index comes from blockIdx.y (SGPR, no integer
// division on the VALU path). 64 threads per pillar (1 thread = 1 feature);
// each wave32 belongs to a single pillar, so the contains-mask branch is
// wave-uniform (whole wave exits, zero divergence).
// Accumulation uses the native CDNA5 global_atomic_add_f32 at device scope
// (non-returning -> STOREcnt; RMW resolved in the 192MB global L2, which
// holds the entire 191MB output, so duplicate-coordinate collisions stay
// on-chip).
// ---------------------------------------------------------------------------
__global__ void __launch_bounds__(256)
pseudoimg_scatter_kernel(const float* __restrict__ pillars,
                         const int*   __restrict__ coord,
                         const int*   __restrict__ contains,
                         float*       __restrict__ out,
                         int P) {
    const int b       = blockIdx.y;                       // batch (SGPR)
    const int t       = blockIdx.x * blockDim.x + threadIdx.x;
    const int p_local = t >> 6;                           // pillar within batch
    const int f       = t & 63;                           // feature
    if (p_local >= P) return;

    const int pillar = b * P + p_local;                   // flat [B*P) index

    // Wave-uniform predicate: whole wave exits together when pillar is empty.
    if (contains[pillar] != 1) return;

    const int x = coord[pillar * 3 + 1];
    const int y = coord[pillar * 3 + 2];

    const float v = pillars[(size_t)pillar * FDIM + f];   // coalesced b32 load

    // out[b][f][x][y], F on the slow axis (reference's permute(0,3,1,2))
    const size_t off =
        (((size_t)(b * FDIM + f) * XSIZE) + (size_t)x) * YSIZE + (size_t)y;
    float* addr = out + off;

    // Non-returning FP32 atomic add, device scope, guaranteed native
    // (avoids any CAS-loop fallback for float atomicAdd).
    asm volatile("global_atomic_add_f32 %0, %1, off scope:SCOPE_DEV"
                 :
                 : "v"(addr), "v"(v)
                 : "memory");
}

// ---------------------------------------------------------------------------
// Launch
// ---------------------------------------------------------------------------
extern "C" void kernel_launch(void* const* d_in, const int* in_sizes, int n_in,
                              void* d_out, int out_size, void* d_ws, size_t ws_size,
                              hipStream_t stream) {
    const float* pillars  = (const float*)d_in[0];   // [B, P, F] f32
    const int*   coord    = (const int*)  d_in[1];   // [B, P, 3] i32 (z,x,y)
    const int*   contains = (const int*)  d_in[2];   // [B, P, 1] i32
    float*       out      = (float*)d_out;           // [B, F, X, Y] f32

    const int BP = in_sizes[2];                      // B*P (contains: 1 elem/pillar)
    const int B  = out_size / (FDIM * XSIZE * YSIZE);
    const int P  = BP / (B > 0 ? B : 1);

    // 1) zero output (out_size = B*64*432*432, divisible by 4)
    long n4 = (long)out_size / 4;
    int  zb = (int)((n4 + 255) / 256);
    pseudoimg_zero_kernel<<<zb, 256, 0, stream>>>((float4*)out, n4);

    // 2) scatter-add: grid.x covers one batch's P*64 threads, grid.y = batch
    int per_batch = P * FDIM;
    int sbx       = (per_batch + 255) / 256;
    dim3 sgrid(sbx, (unsigned)B, 1);
    pseudoimg_scatter_kernel<<<sgrid, 256, 0, stream>>>(pillars, coord, contains,
                                                        out, P);
}